// EdgeConv_5669356835846
// MI455X (gfx1250) — compile-verified
//
#include <hip/hip_runtime.h>

typedef __attribute__((ext_vector_type(2))) float v2f;
typedef __attribute__((ext_vector_type(8))) float v8f;

#define NPTS 4096
#define CCH  64
#define KNN  20
#define FMAX 3.402823466e+38f

// ---------------------------------------------------------------------------
// ws layout (bytes):
//   norms : 4*4096 floats            @ 0          (65536)
//   idx   : 4*4096*20 ints           @ 65536      (1310720)
//   y     : 4*4096*64 floats         @ 1376256    (4194304)   y = X * W1^T
//   z     : 4*4096*64 floats         @ 5570560    (4194304)   z = X * (W2-W1)^T
// ---------------------------------------------------------------------------

__global__ void norms_kernel(const float* __restrict__ x, float* __restrict__ norms) {
    int p = blockIdx.x * blockDim.x + threadIdx.x;   // 0..16383
    int b = p >> 12, n = p & 4095;
    const float* xb = x + b * CCH * NPTS + n;
    float s = 0.f;
    #pragma unroll
    for (int c = 0; c < CCH; ++c) { float v = xb[c * NPTS]; s += v * v; }
    norms[p] = s;
}

// One wave (32 threads) owns 16 rows; sweeps all 4096 columns in 16-wide tiles.
// Gram tile via 16x chained v_wmma_f32_16x16x4_f32 (K=64), then streaming top-K.
__global__ void __launch_bounds__(32) knn_kernel(const float* __restrict__ x,
                                                 const float* __restrict__ norms,
                                                 int* __restrict__ knn_idx) {
    __shared__ float distT[16][17];          // padded: conflict-free row scans
    __shared__ float kd[16 * KNN];
    __shared__ int   ki[16 * KNN];

    const int lane = threadIdx.x;
    const int half = lane >> 4;              // 0/1 : K-split per WMMA A/B layout
    const int lm   = lane & 15;
    const int b    = blockIdx.y;
    const int rowBase = blockIdx.x * 16;
    const float* xb = x + b * CCH * NPTS;
    const float* nb = norms + b * NPTS;

    // A fragments: 16 rows x 64 channels, frag k covers c = 4k..4k+3
    v2f a[16];
    #pragma unroll
    for (int k = 0; k < 16; ++k) {
        int c = 4 * k + 2 * half;
        a[k].x = xb[(c    ) * NPTS + rowBase + lm];
        a[k].y = xb[(c + 1) * NPTS + rowBase + lm];
    }
    float rn[8];
    #pragma unroll
    for (int r = 0; r < 8; ++r) rn[r] = nb[rowBase + r + 8 * half];

    if (lane < 16) {
        for (int j = 0; j < KNN; ++j) { kd[lane * KNN + j] = FMAX; ki[lane * KNN + j] = 0; }
    }
    float curMax = FMAX; int curArg = 0;
    __syncthreads();

    for (int t = 0; t < NPTS / 16; ++t) {
        const int colBase = t * 16;

        // B fragments: same pattern as A (Gram matrix => B is transpose-load of X)
        v2f bb[16];
        #pragma unroll
        for (int k = 0; k < 16; ++k) {
            int c = 4 * k + 2 * half;
            bb[k].x = xb[(c    ) * NPTS + colBase + lm];
            bb[k].y = xb[(c + 1) * NPTS + colBase + lm];
        }

        v8f acc = {};
        #pragma unroll
        for (int k = 0; k < 16; ++k)
            acc = __builtin_amdgcn_wmma_f32_16x16x4_f32(false, a[k], false, bb[k],
                                                        (short)0, acc, false, false);

        float cn = nb[colBase + lm];
        #pragma unroll
        for (int r = 0; r < 8; ++r)
            distT[r + 8 * half][lm] = rn[r] + cn - 2.0f * acc[r];
        __syncthreads();

        if (lane < 16) {                     // lane owns row (rowBase+lane)
            for (int j = 0; j < 16; ++j) {
                float d = distT[lane][j];
                if (d < curMax) {            // O(1) reject for non-candidates
                    kd[lane * KNN + curArg] = d;
                    ki[lane * KNN + curArg] = colBase + j;
                    float m = -FMAX; int arg = 0;
                    for (int q = 0; q < KNN; ++q) {
                        float v = kd[lane * KNN + q];
                        if (v > m) { m = v; arg = q; }
                    }
                    curMax = m; curArg = arg;
                }
            }
        }
        __syncthreads();
    }

    if (lane < 16) {
        int* op = knn_idx + (b * NPTS + rowBase + lane) * KNN;
        for (int j = 0; j < KNN; ++j) op[j] = ki[lane * KNN + j];
    }
}

// y = X*W1^T, z = X*(W2-W1)^T ; 16 rows per wave, 4 column tiles of 16 outputs.
__global__ void __launch_bounds__(32) proj_kernel(const float* __restrict__ x,
                                                  const float* __restrict__ W,
                                                  float* __restrict__ y,
                                                  float* __restrict__ z) {
    const int lane = threadIdx.x;
    const int half = lane >> 4;
    const int lm   = lane & 15;
    const int b    = blockIdx.y;
    const int rowBase = blockIdx.x * 16;
    const float* xb = x + b * CCH * NPTS;

    v2f a[16];
    #pragma unroll
    for (int k = 0; k < 16; ++k) {
        int c = 4 * k + 2 * half;
        a[k].x = xb[(c    ) * NPTS + rowBase + lm];
        a[k].y = xb[(c + 1) * NPTS + rowBase + lm];
    }

    for (int ot = 0; ot < 4; ++ot) {
        const int o = ot * 16 + lm;          // output channel for this lane
        v8f accY = {}, accZ = {};
        #pragma unroll
        for (int k = 0; k < 16; ++k) {
            int c = 4 * k + 2 * half;
            float w0 = W[o * 128 + c], w1 = W[o * 128 + c + 1];
            v2f bw, bz;
            bw.x = w0;
            bw.y = w1;
            bz.x = W[o * 128 + 64 + c    ] - w0;
            bz.y = W[o * 128 + 64 + c + 1] - w1;
            accY = __builtin_amdgcn_wmma_f32_16x16x4_f32(false, a[k], false, bw,
                                                         (short)0, accY, false, false);
            accZ = __builtin_amdgcn_wmma_f32_16x16x4_f32(false, a[k], false, bz,
                                                         (short)0, accZ, false, false);
        }
        #pragma unroll
        for (int r = 0; r < 8; ++r) {
            int row = rowBase + r + 8 * half;
            y[(b * NPTS + row) * CCH + o] = accY[r];
            z[(b * NPTS + row) * CCH + o] = accZ[r];
        }
    }
}

// out[b,o,n] = z[b,n,o] + max_k y[b, idx[b,n,k], o]
__global__ void gathermax_kernel(const float* __restrict__ y,
                                 const float* __restrict__ z,
                                 const int* __restrict__ knn_idx,
                                 float* __restrict__ out) {
    int t = blockIdx.x * blockDim.x + threadIdx.x;   // 0 .. 16384*64-1
    int o = t & 63;
    int p = t >> 6;
    int b = p >> 12, n = p & 4095;
    const int*   ip = knn_idx + p * KNN;
    const float* yb = y + b * NPTS * CCH;
    float best = -FMAX;
    #pragma unroll
    for (int k = 0; k < KNN; ++k) {
        int m = ip[k];
        best = fmaxf(best, yb[m * CCH + o]);
    }
    best += z[p * CCH + o];
    out[(b * CCH + o) * NPTS + n] = best;
}

extern "C" void kernel_launch(void* const* d_in, const int* in_sizes, int n_in,
                              void* d_out, int out_size, void* d_ws, size_t ws_size,
                              hipStream_t stream) {
    (void)in_sizes; (void)n_in; (void)out_size; (void)ws_size;
    const float* x = (const float*)d_in[0];
    const float* W = (const float*)d_in[1];
    float* out = (float*)d_out;

    char* ws = (char*)d_ws;
    float* norms = (float*)(ws);
    int*   idxb  = (int*)  (ws + 65536);
    float* y     = (float*)(ws + 65536 + 1310720);
    float* z     = (float*)(ws + 65536 + 1310720 + 4194304);

    norms_kernel    <<<dim3(64), 256, 0, stream>>>(x, norms);
    proj_kernel     <<<dim3(256, 4), 32, 0, stream>>>(x, W, y, z);
    knn_kernel      <<<dim3(256, 4), 32, 0, stream>>>(x, norms, idxb);
    gathermax_kernel<<<dim3(4096), 256, 0, stream>>>(y, z, idxb, out);
}